// VSAMemory_57458072486501
// MI455X (gfx1250) — compile-verified
//
#include <hip/hip_runtime.h>
#include <math.h>

// ---------------- problem constants (match reference) ----------------
#define DIMX     1024
#define NSLOTS   32768
#define BUCKETS  128
#define KTOP     16
#define BATCHN   2048
#define SPB      256      // slots per bucket

// Only check the builtin in the *device* pass; the host pass targets x86 and
// reports __has_builtin(amdgcn builtin) == 0 even though device code may use it.
#if defined(__HIP_DEVICE_COMPILE__)
#if !__has_builtin(__builtin_amdgcn_wmma_f32_16x16x4_f32)
#error "gfx1250 device pass lacks __builtin_amdgcn_wmma_f32_16x16x4_f32"
#endif
#endif

typedef __attribute__((ext_vector_type(2))) float v2f;
typedef __attribute__((ext_vector_type(8))) float v8f;

__device__ __forceinline__ v8f wmma4_f32(v2f a, v2f b, v8f c) {
  // D = A(16x4) * B(4x16) + C(16x16), full fp32
  return __builtin_amdgcn_wmma_f32_16x16x4_f32(false, a, false, b, (short)0, c,
                                               false, false);
}

// ---------------- output layout (floats, concatenated in return order) ----
constexpr size_t O_OUTS = 0;                                // [BATCH, DIM]
constexpr size_t O_BKT  = (size_t)BATCHN * DIMX;            // [BATCH]
constexpr size_t O_ENT  = O_BKT + BATCHN;                   // [BATCH]
constexpr size_t O_IDX  = O_ENT + BATCHN;                   // [BATCH, KTOP]

// ---------------- workspace layout (bytes) ----------------
constexpr size_t WS_KEYN   = 0;                                     // BATCH*DIM f32
constexpr size_t WS_KNORM  = WS_KEYN   + (size_t)BATCHN * DIMX * 4; // BATCH f32
constexpr size_t WS_ANORM  = WS_KNORM  + (size_t)BATCHN * 4;        // NSLOTS f32
constexpr size_t WS_SCORES = WS_ANORM  + (size_t)NSLOTS * 4;        // BATCH*BUCKETS f32
constexpr size_t WS_BID    = WS_SCORES + (size_t)BATCHN * BUCKETS * 4; // BATCH i32
constexpr size_t WS_BCNT   = WS_BID    + (size_t)BATCHN * 4;        // BUCKETS i32
constexpr size_t WS_BOFF   = WS_BCNT   + (size_t)BUCKETS * 4;       // BUCKETS i32
constexpr size_t WS_BFILL  = WS_BOFF   + (size_t)BUCKETS * 4;       // BUCKETS i32
constexpr size_t WS_ORD    = WS_BFILL  + (size_t)BUCKETS * 4;       // BATCH i32
constexpr size_t WS_SIMS   = WS_ORD    + (size_t)BATCHN * 4;        // BATCH*SPB f32
constexpr size_t WS_TIDX   = WS_SIMS   + (size_t)BATCHN * SPB * 4;  // BATCH*KTOP i32
constexpr size_t WS_CONT   = WS_TIDX   + (size_t)BATCHN * KTOP * 4; // BATCH*DIM f32

// ---------------- reduction helpers (wave32) ----------------
__device__ __forceinline__ float waveSum(float v) {
  for (int off = 16; off > 0; off >>= 1) v += __shfl_down(v, off, 32);
  return v;
}

__device__ __forceinline__ float blockSum256(float v, float* red8) {
  int lane = threadIdx.x & 31, w = threadIdx.x >> 5;
  v = waveSum(v);
  if (lane == 0) red8[w] = v;
  __syncthreads();
  if (threadIdx.x == 0) {
    float t = 0.f;
    for (int i = 0; i < 8; i++) t += red8[i];
    red8[0] = t;
  }
  __syncthreads();
  float r = red8[0];
  __syncthreads();
  return r;
}

// ---------------- K0: address row norms ----------------
__global__ void k_addr_norm(const float* __restrict__ addresses,
                            float* __restrict__ anorm) {
  __shared__ float red[8];
  int r = blockIdx.x;
  const float4* row = (const float4*)(addresses + (size_t)r * DIMX);
  float4 v = row[threadIdx.x];
  float ss = v.x * v.x + v.y * v.y + v.z * v.z + v.w * v.w;
  ss = blockSum256(ss, red);
  if (threadIdx.x == 0) anorm[r] = sqrtf(ss);
}

// ---------------- K1: normalize keys ----------------
__global__ void k_norm_keys(const float* __restrict__ keys,
                            float* __restrict__ keyn,
                            float* __restrict__ knorm) {
  __shared__ float red[8];
  int b = blockIdx.x;
  const float4* row = (const float4*)(keys + (size_t)b * DIMX);
  float4 v = row[threadIdx.x];
  float ss = v.x * v.x + v.y * v.y + v.z * v.z + v.w * v.w;
  ss = blockSum256(ss, red);
  float n = sqrtf(ss);
  float inv = 1.0f / (n + 1e-8f);
  float4 o = make_float4(v.x * inv, v.y * inv, v.z * inv, v.w * inv);
  ((float4*)(keyn + (size_t)b * DIMX))[threadIdx.x] = o;
  if (threadIdx.x == 0) knorm[b] = n * inv;  // == ||key_n||
}

// ---------------- K2: scores = key_n @ projT via f32 WMMA 16x16x4 --------
// grid.x = BATCH/16 (M tiles), block = 256 = 8 waves, wave w -> N tile w*16
__global__ void k_scores(const float* __restrict__ keyn,
                         const float* __restrict__ proj,
                         float* __restrict__ scores) {
  int mBase = blockIdx.x * 16;
  int wave = threadIdx.x >> 5;
  int lane = threadIdx.x & 31;
  int half = lane >> 4;   // 0: K={k,k+1}, 1: K={k+2,k+3}
  int l16  = lane & 15;
  int nBase = wave * 16;

  const float* ar = keyn + (size_t)(mBase + l16) * DIMX + 2 * half;
  const float* br = proj + (size_t)(nBase + l16) * DIMX + 2 * half;

  v8f acc = {};
#pragma unroll 8
  for (int k = 0; k < DIMX; k += 4) {
    float2 a2 = *(const float2*)(ar + k);
    float2 b2 = *(const float2*)(br + k);
    v2f av; av.x = a2.x; av.y = a2.y;
    v2f bv; bv.x = b2.x; bv.y = b2.y;
    acc = wmma4_f32(av, bv, acc);
  }
#pragma unroll
  for (int v = 0; v < 8; v++)
    scores[(size_t)(mBase + v + half * 8) * BUCKETS + nBase + l16] = acc[v];
}

// ---------------- K3a: zero bucket counters ----------------
__global__ void k_zero(int* __restrict__ bcnt, int* __restrict__ bfill) {
  bcnt[threadIdx.x] = 0;
  bfill[threadIdx.x] = 0;
}

// ---------------- K3b: per-row argmax over 128 buckets + histogram -------
__global__ void k_argmax(const float* __restrict__ scores,
                         int* __restrict__ bid, int* __restrict__ bcnt,
                         float* __restrict__ dout) {
  __shared__ float sval[128];
  __shared__ int   sidx[128];
  int b = blockIdx.x, t = threadIdx.x;
  sval[t] = scores[(size_t)b * BUCKETS + t];
  sidx[t] = t;
  __syncthreads();
  for (int off = 64; off > 0; off >>= 1) {
    if (t < off) {
      float ov = sval[t + off];
      int   oi = sidx[t + off];
      if (ov > sval[t] || (ov == sval[t] && oi < sidx[t])) {
        sval[t] = ov; sidx[t] = oi;   // first-occurrence tie break
      }
    }
    __syncthreads();
  }
  if (t == 0) {
    int g = sidx[0];
    bid[b] = g;
    dout[O_BKT + b] = (float)g;
    atomicAdd(&bcnt[g], 1);
  }
}

// ---------------- K3c: exclusive prefix sum over 128 counts --------------
__global__ void k_prefix(const int* __restrict__ bcnt, int* __restrict__ boff,
                         int* __restrict__ bfill) {
  __shared__ int sc[128];
  int t = threadIdx.x;
  int my = bcnt[t];
  sc[t] = my;
  __syncthreads();
  for (int off = 1; off < 128; off <<= 1) {
    int v = (t >= off) ? sc[t - off] : 0;
    __syncthreads();
    sc[t] += v;
    __syncthreads();
  }
  boff[t] = sc[t] - my;   // exclusive
  bfill[t] = 0;
}

// ---------------- K3d: scatter batch ids into bucket-ordered list --------
__global__ void k_scatter(const int* __restrict__ bid,
                          const int* __restrict__ boff,
                          int* __restrict__ bfill, int* __restrict__ ord) {
  int b = blockIdx.x * blockDim.x + threadIdx.x;
  int g = bid[b];
  int pos = atomicAdd(&bfill[g], 1);
  ord[boff[g] + pos] = b;
}

// ---------------- K4: bucketed cosine sims via f32 WMMA ------------------
// grid = (SPB/16 slot tiles, BUCKETS), block = 32 (one wave)
__global__ void k_sims(const float* __restrict__ keyn,
                       const float* __restrict__ addresses,
                       const float* __restrict__ knorm,
                       const float* __restrict__ anorm,
                       const int* __restrict__ ord,
                       const int* __restrict__ boff,
                       const int* __restrict__ bcnt,
                       float* __restrict__ sims) {
  int t = blockIdx.x;        // slot tile
  int g = blockIdx.y;        // bucket
  int lane = threadIdx.x & 31;
  int half = lane >> 4;
  int l16  = lane & 15;

  int cnt = bcnt[g];
  int off = boff[g];
  int slotN = (t * 16 + l16) * BUCKETS + g;   // this lane's N-column slot id
  const float* br = addresses + (size_t)slotN * DIMX + 2 * half;
  float dA = fmaxf(anorm[slotN], 1e-8f);

  for (int base = 0; base < cnt; base += 16) {
    int ridx = base + l16;
    int bm = ord[off + (ridx < cnt ? ridx : cnt - 1)];  // pad by clamping
    const float* ar = keyn + (size_t)bm * DIMX + 2 * half;

    v8f acc = {};
#pragma unroll 8
    for (int k = 0; k < DIMX; k += 4) {
      float2 a2 = *(const float2*)(ar + k);
      float2 b2 = *(const float2*)(br + k);
      v2f av; av.x = a2.x; av.y = a2.y;
      v2f bv; bv.x = b2.x; bv.y = b2.y;
      acc = wmma4_f32(av, bv, acc);
    }

#pragma unroll
    for (int v = 0; v < 8; v++) {
      int M = v + half * 8;
      int rowi = base + M;
      if (rowi < cnt) {
        int b = ord[off + rowi];
        float den = dA * fmaxf(knorm[b], 1e-8f);
        sims[(size_t)b * SPB + t * 16 + l16] = acc[v] / den;
      }
    }
  }
}

// ---------------- K5: iterative wave top-16 + entropy + idx --------------
// grid = BATCH, block = 32
__global__ void k_topk(const float* __restrict__ sims,
                       const int* __restrict__ bid,
                       int* __restrict__ tidx, float* __restrict__ dout) {
  int b = blockIdx.x;
  int lane = threadIdx.x;
  float vals[8];
#pragma unroll
  for (int j = 0; j < 8; j++)
    vals[j] = sims[(size_t)b * SPB + j * 32 + lane];

  float tv[KTOP];
  int   ti[KTOP];
#pragma unroll
  for (int it = 0; it < KTOP; it++) {
    float bv = vals[0];
    int   bi = lane;                       // s = j*32 + lane
#pragma unroll
    for (int j = 1; j < 8; j++) {
      if (vals[j] > bv) { bv = vals[j]; bi = j * 32 + lane; }
    }
    for (int o = 16; o > 0; o >>= 1) {
      float ov = __shfl_down(bv, o, 32);
      int   oi = __shfl_down(bi, o, 32);
      if (ov > bv || (ov == bv && oi < bi)) { bv = ov; bi = oi; }
    }
    bv = __shfl(bv, 0, 32);
    bi = __shfl(bi, 0, 32);
    tv[it] = bv;
    ti[it] = bi;
    if (lane == (bi & 31)) {
#pragma unroll
      for (int j = 0; j < 8; j++)
        if (j == (bi >> 5)) vals[j] = -INFINITY;
    }
  }

  int g = bid[b];
  if (lane < KTOP) {
    int gi = ti[lane] * BUCKETS + g;      // global slot id
    tidx[(size_t)b * KTOP + lane] = gi;
    dout[O_IDX + (size_t)b * KTOP + lane] = (float)gi;
  }
  if (lane == 0) {
    float m = tv[0];
#pragma unroll
    for (int i = 1; i < KTOP; i++) m = fmaxf(m, tv[i]);
    float s = 0.f, p[KTOP];
#pragma unroll
    for (int i = 0; i < KTOP; i++) { p[i] = __expf(tv[i] - m); s += p[i]; }
    float ent = 0.f;
#pragma unroll
    for (int i = 0; i < KTOP; i++) {
      float pr = p[i] / s;
      ent -= pr * __logf(fmaxf(pr, 1e-9f));
    }
    dout[O_ENT + b] = ent;
  }
}

// ---------------- K6: content = sum of 16 gathered memory rows -----------
__global__ void k_content(const float* __restrict__ memory,
                          const int* __restrict__ tidx,
                          float* __restrict__ content) {
  __shared__ int sidx[KTOP];
  int b = blockIdx.x, t = threadIdx.x;
  if (t < KTOP) sidx[t] = tidx[(size_t)b * KTOP + t];
  __syncthreads();
  float ax = 0.f, ay = 0.f, az = 0.f, aw = 0.f;
  for (int r = 0; r < KTOP; r++) {
    if (r + 1 < KTOP)
      __builtin_prefetch(memory + (size_t)sidx[r + 1] * DIMX + t * 4, 0, 1);
    const float4* row = (const float4*)(memory + (size_t)sidx[r] * DIMX);
    float4 v = row[t];
    ax += v.x; ay += v.y; az += v.z; aw += v.w;
  }
  ((float4*)(content + (size_t)b * DIMX))[t] =
      make_float4(ax, ay, az, aw);
}

// ---------------- K7: HRR unbind (rfft/irfft) + normalize ----------------
// Packed real trick: one 1024-pt complex Stockham FFT of (key_n + i*content)
// gives both rffts; Hermitian inverse gives b_hat. block = 256.
__global__ void k_fft_unbind(const float* __restrict__ keyn,
                             const float* __restrict__ content,
                             float* __restrict__ dout) {
  __shared__ float2 bufA[DIMX];
  __shared__ float2 bufB[DIMX];
  __shared__ float red[8];
  int b = blockIdx.x, tid = threadIdx.x;
  const float* kr = keyn + (size_t)b * DIMX;
  const float* cr = content + (size_t)b * DIMX;

  for (int i = tid; i < DIMX; i += 256)
    bufA[i] = make_float2(kr[i], cr[i]);
  __syncthreads();

  float2* src = bufA;
  float2* dst = bufB;

  // forward FFT (sign = -1), radix-2 Stockham autosort, 10 stages
  for (int s = 0; s < 10; s++) {
    int L = 1 << s;
    for (int t = tid; t < DIMX / 2; t += 256) {
      int p = t & (L - 1);
      int q = t >> s;
      float ang = -3.14159265358979323846f * (float)p / (float)L;
      float sn, cs;
      __sincosf(ang, &sn, &cs);
      float2 a = src[t];
      float2 bb = src[t + DIMX / 2];
      float2 wb = make_float2(cs * bb.x - sn * bb.y, cs * bb.y + sn * bb.x);
      int o = (q << (s + 1)) + p;
      dst[o]     = make_float2(a.x + wb.x, a.y + wb.y);
      dst[o + L] = make_float2(a.x - wb.x, a.y - wb.y);
    }
    __syncthreads();
    float2* tmp = src; src = dst; dst = tmp;
  }
  // src = Z = FFT(key + i*content). Build G spectrum in dst (Hermitian).
  for (int k = tid; k <= DIMX / 2; k += 256) {
    float2 Zk = src[k];
    float2 Zn = src[(DIMX - k) & (DIMX - 1)];
    float Ar = 0.5f * (Zk.x + Zn.x);
    float Ai = 0.5f * (Zk.y - Zn.y);
    float Cr = 0.5f * (Zk.y + Zn.y);
    float Ci = -0.5f * (Zk.x - Zn.x);
    float d = Ar * Ar + Ai * Ai + 1e-8f;
    float Gr = (Cr * Ar + Ci * Ai) / d;    // C * conj(A) / (|A|^2 + eps)
    float Gi = (Ci * Ar - Cr * Ai) / d;
    dst[k] = make_float2(Gr, Gi);
    if (k > 0 && k < DIMX / 2) dst[DIMX - k] = make_float2(Gr, -Gi);
  }
  __syncthreads();
  { float2* tmp = src; src = dst; dst = tmp; }   // src = G

  // inverse FFT (sign = +1)
  for (int s = 0; s < 10; s++) {
    int L = 1 << s;
    for (int t = tid; t < DIMX / 2; t += 256) {
      int p = t & (L - 1);
      int q = t >> s;
      float ang = 3.14159265358979323846f * (float)p / (float)L;
      float sn, cs;
      __sincosf(ang, &sn, &cs);
      float2 a = src[t];
      float2 bb = src[t + DIMX / 2];
      float2 wb = make_float2(cs * bb.x - sn * bb.y, cs * bb.y + sn * bb.x);
      int o = (q << (s + 1)) + p;
      dst[o]     = make_float2(a.x + wb.x, a.y + wb.y);
      dst[o + L] = make_float2(a.x - wb.x, a.y - wb.y);
    }
    __syncthreads();
    float2* tmp = src; src = dst; dst = tmp;
  }

  // b_hat = Re(src)/N, then L2-normalize with eps
  float vals[4];
  float ss = 0.f;
#pragma unroll
  for (int j = 0; j < 4; j++) {
    int i = tid + j * 256;
    float v = src[i].x * (1.0f / (float)DIMX);
    vals[j] = v;
    ss += v * v;
  }
  ss = blockSum256(ss, red);
  float inv = 1.0f / (sqrtf(ss) + 1e-8f);
#pragma unroll
  for (int j = 0; j < 4; j++)
    dout[O_OUTS + (size_t)b * DIMX + tid + j * 256] = vals[j] * inv;
}

// ---------------- launch ----------------
extern "C" void kernel_launch(void* const* d_in, const int* in_sizes, int n_in,
                              void* d_out, int out_size, void* d_ws,
                              size_t ws_size, hipStream_t stream) {
  (void)in_sizes; (void)n_in; (void)out_size; (void)ws_size;
  const float* keys      = (const float*)d_in[0];
  const float* memory    = (const float*)d_in[1];
  const float* addresses = (const float*)d_in[2];
  const float* proj      = (const float*)d_in[3];
  float* dout = (float*)d_out;

  char* ws = (char*)d_ws;
  float* keyn   = (float*)(ws + WS_KEYN);
  float* knorm  = (float*)(ws + WS_KNORM);
  float* anorm  = (float*)(ws + WS_ANORM);
  float* scores = (float*)(ws + WS_SCORES);
  int*   bid    = (int*)  (ws + WS_BID);
  int*   bcnt   = (int*)  (ws + WS_BCNT);
  int*   boff   = (int*)  (ws + WS_BOFF);
  int*   bfill  = (int*)  (ws + WS_BFILL);
  int*   ord    = (int*)  (ws + WS_ORD);
  float* sims   = (float*)(ws + WS_SIMS);
  int*   tidx   = (int*)  (ws + WS_TIDX);
  float* cont   = (float*)(ws + WS_CONT);

  k_addr_norm<<<NSLOTS, 256, 0, stream>>>(addresses, anorm);
  k_norm_keys<<<BATCHN, 256, 0, stream>>>(keys, keyn, knorm);
  k_scores<<<BATCHN / 16, 256, 0, stream>>>(keyn, proj, scores);
  k_zero<<<1, BUCKETS, 0, stream>>>(bcnt, bfill);
  k_argmax<<<BATCHN, BUCKETS, 0, stream>>>(scores, bid, bcnt, dout);
  k_prefix<<<1, BUCKETS, 0, stream>>>(bcnt, boff, bfill);
  k_scatter<<<BATCHN / 256, 256, 0, stream>>>(bid, boff, bfill, ord);
  k_sims<<<dim3(SPB / 16, BUCKETS), 32, 0, stream>>>(keyn, addresses, knorm,
                                                     anorm, ord, boff, bcnt,
                                                     sims);
  k_topk<<<BATCHN, 32, 0, stream>>>(sims, bid, tidx, dout);
  k_content<<<BATCHN, 256, 0, stream>>>(memory, tidx, cont);
  k_fft_unbind<<<BATCHN, 256, 0, stream>>>(keyn, cont, dout);
}